// HopfRNNCellTheta_80221399154860
// MI455X (gfx1250) — compile-verified
//
#include <hip/hip_runtime.h>
#include <hip/hip_bf16.h>

#define NN 8192
#define UNITS 33
#define SPLITS 8
#define KCH (NN / SPLITS)        // 1024 k-values per split
#define CHUNK_KC 8               // 8 k-chunks (of 4) per LDS stage = 32 k-values
#define CHUNKS (KCH / (4 * CHUNK_KC))   // 32 staging chunks per split
#define PACK_FLOATS_PER_KCOUT 4096      // one stage = 8 kc * 32 lanes * 16 floats

typedef float v2f __attribute__((ext_vector_type(2)));
typedef float v8f __attribute__((ext_vector_type(8)));
typedef int   v4i __attribute__((vector_size(16)));   // matches builtin param type

#if defined(__has_builtin)
#if __has_builtin(__builtin_amdgcn_global_load_async_to_lds_b128) && \
    __has_builtin(__builtin_amdgcn_s_wait_asynccnt)
#define USE_ASYNC_LDS 1
#endif
#endif

#define AS1 __attribute__((address_space(1)))
#define AS3 __attribute__((address_space(3)))

__device__ __forceinline__ v8f wmma_f32(v2f a, v2f b, v8f c) {
    return __builtin_amdgcn_wmma_f32_16x16x4_f32(
        false, a, false, b, (short)0, c, false, false);
}

// ---------------------------------------------------------------------------
// Pack RHS (z_n and x panels, re/im) into the exact per-lane WMMA B-operand
// layout, grouped so one 32-k chunk is a contiguous 16 KB block:
//   pk[kcOuter][slot j=0..3][kcl=0..7][lane=0..31] : float4
//   j=0:(zr0,zr1) j=1:(zi0,zi1) j=2:(xr0,xr1) j=3:(xi0,xi1)
// Lane: n = lane&15, K pair = 2*(lane>>4), 2*(lane>>4)+1 within the k-chunk.
// ---------------------------------------------------------------------------
__global__ __launch_bounds__(256) void hopf_pack_kernel(
    const float* __restrict__ z_re, const float* __restrict__ z_im,
    const float* __restrict__ x_re, const float* __restrict__ x_im,
    float* __restrict__ pk)
{
    const int tid = blockIdx.x * 256 + threadIdx.x;   // 2048*32 threads
    const int kc   = tid >> 5;        // k-chunk (4 k-values), 0..2047
    const int lane = tid & 31;
    const int m  = lane & 15;
    const int hk = (lane >> 4) * 2;
    const int k0 = kc * 4 + hk;
    const int k1 = k0 + 1;

    const int kcOuter = kc >> 3;
    const int kcl     = kc & 7;
    float* base = pk + (size_t)kcOuter * PACK_FLOATS_PER_KCOUT + kcl * 128 + lane * 4;

    float4 v0, v1, v2, v3;
    v0.x = z_re[k0 * UNITS + 1 + m];  v0.y = z_re[k1 * UNITS + 1 + m];
    v0.z = z_re[k0 * UNITS + 17 + m]; v0.w = z_re[k1 * UNITS + 17 + m];
    v1.x = z_im[k0 * UNITS + 1 + m];  v1.y = z_im[k1 * UNITS + 1 + m];
    v1.z = z_im[k0 * UNITS + 17 + m]; v1.w = z_im[k1 * UNITS + 17 + m];
    v2.x = x_re[k0 * 32 + m];         v2.y = x_re[k1 * 32 + m];
    v2.z = x_re[k0 * 32 + 16 + m];    v2.w = x_re[k1 * 32 + 16 + m];
    v3.x = x_im[k0 * 32 + m];         v3.y = x_im[k1 * 32 + m];
    v3.z = x_im[k0 * 32 + 16 + m];    v3.w = x_im[k1 * 32 + 16 + m];

    *(float4*)(base)        = v0;
    *(float4*)(base + 1024) = v1;
    *(float4*)(base + 2048) = v2;
    *(float4*)(base + 3072) = v3;
}

// ---------------------------------------------------------------------------
// GEMM: block = 8 waves sharing one K-split; each wave owns a 16-row tile and
// all 32 output columns. RHS streamed via async global->LDS; A streamed as
// one b64/matrix/k-step (the HBM-bound 1.07 GB read).
// ---------------------------------------------------------------------------
__global__ __launch_bounds__(256) void hopf_gemm_kernel(
    const float* __restrict__ A_re, const float* __restrict__ A_im,
    const float* __restrict__ B_re, const float* __restrict__ B_im,
    const float* __restrict__ pk,
    float* __restrict__ pRe, float* __restrict__ pIm)
{
    __shared__ __align__(16) float sRHS[PACK_FLOATS_PER_KCOUT];  // 16 KB

    const int t     = threadIdx.x;
    const int lane  = t & 31;
    const int wave  = t >> 5;
    const int split = blockIdx.x >> 6;                 // 0..7  (same for block)
    const int tile  = (blockIdx.x & 63) * 8 + wave;    // 0..511
    const int rowBase = tile * 16;
    const int kStart  = split * KCH;

    const int m  = lane & 15;
    const int hk = (lane >> 4) * 2;

    v8f accRP0 = {}, accRP1 = {};
    v8f accRN0 = {}, accRN1 = {};
    v8f accI0  = {}, accI1  = {};

    unsigned aOff = (unsigned)((rowBase + m) * NN + kStart + hk);
    const float* __restrict__ src = pk + (size_t)(kStart / 32) * PACK_FLOATS_PER_KCOUT;

    for (int c = 0; c < CHUNKS; ++c) {
        __syncthreads();   // previous chunk fully consumed before overwrite
        // ---- stage 16 KB of packed RHS into LDS (4 x b128 per thread) ----
        #pragma unroll
        for (int j = 0; j < 4; ++j) {
            const int fIdx = (j * 256 + t) * 4;
#ifdef USE_ASYNC_LDS
            __builtin_amdgcn_global_load_async_to_lds_b128(
                (AS1 v4i*)(src + fIdx), (AS3 v4i*)(&sRHS[fIdx]), 0, 0);
#else
            *(float4*)(&sRHS[fIdx]) = *(const float4*)(src + fIdx);
#endif
        }
#ifdef USE_ASYNC_LDS
        __builtin_amdgcn_s_wait_asynccnt(0);
#endif
        __syncthreads();

        // ---- 8 k-chunks of 4 from this stage ----
        #pragma unroll
        for (int kcl = 0; kcl < CHUNK_KC; ++kcl) {
            const v2f a_re = *(const v2f*)(A_re + aOff + 4 * kcl);
            const v2f a_im = *(const v2f*)(A_im + aOff + 4 * kcl);
            const v2f b_re = *(const v2f*)(B_re + aOff + 4 * kcl);
            const v2f b_im = *(const v2f*)(B_im + aOff + 4 * kcl);

            const int lbase = kcl * 128 + lane * 4;   // conflict-free: 16B lane stride
            const float4 q0 = *(const float4*)(&sRHS[lbase]);          // zr0|zr1
            const float4 q1 = *(const float4*)(&sRHS[lbase + 1024]);   // zi0|zi1
            const float4 q2 = *(const float4*)(&sRHS[lbase + 2048]);   // xr0|xr1
            const float4 q3 = *(const float4*)(&sRHS[lbase + 3072]);   // xi0|xi1
            const v2f zr0 = {q0.x, q0.y}, zr1 = {q0.z, q0.w};
            const v2f zi0 = {q1.x, q1.y}, zi1 = {q1.z, q1.w};
            const v2f xr0 = {q2.x, q2.y}, xr1 = {q2.z, q2.w};
            const v2f xi0 = {q3.x, q3.y}, xi1 = {q3.z, q3.w};

            accRP0 = wmma_f32(a_re, zr0, accRP0);
            accRP0 = wmma_f32(b_re, xr0, accRP0);
            accRN0 = wmma_f32(a_im, zi0, accRN0);
            accRN0 = wmma_f32(b_im, xi0, accRN0);
            accI0  = wmma_f32(a_re, zi0, accI0);
            accI0  = wmma_f32(a_im, zr0, accI0);
            accI0  = wmma_f32(b_re, xi0, accI0);
            accI0  = wmma_f32(b_im, xr0, accI0);
            accRP1 = wmma_f32(a_re, zr1, accRP1);
            accRP1 = wmma_f32(b_re, xr1, accRP1);
            accRN1 = wmma_f32(a_im, zi1, accRN1);
            accRN1 = wmma_f32(b_im, xi1, accRN1);
            accI1  = wmma_f32(a_re, zi1, accI1);
            accI1  = wmma_f32(a_im, zr1, accI1);
            accI1  = wmma_f32(b_re, xi1, accI1);
            accI1  = wmma_f32(b_im, xr1, accI1);
        }
        aOff += 4 * CHUNK_KC;
        src  += PACK_FLOATS_PER_KCOUT;
    }

    // C/D layout: VGPR r -> M = r + 8*(lane>>4); N = lane&15
    const int nCol = lane & 15;
    const int rOff = (lane >> 4) * 8;
    float* baseRe = pRe + (size_t)split * NN * 32;
    float* baseIm = pIm + (size_t)split * NN * 32;
    #pragma unroll
    for (int r = 0; r < 8; ++r) {
        const int row = rowBase + rOff + r;
        baseRe[row * 32 + nCol]      = accRP0[r] - accRN0[r];
        baseRe[row * 32 + nCol + 16] = accRP1[r] - accRN1[r];
        baseIm[row * 32 + nCol]      = accI0[r];
        baseIm[row * 32 + nCol + 16] = accI1[r];
    }
}

// ---------------------------------------------------------------------------
// Reduce K-splits, complex tanh, passthrough z[:,0], interleaved complex64 out.
// tanh(a+bi) = (sinh 2a + i sin 2b) / (cosh 2a + cos 2b)
// ---------------------------------------------------------------------------
__global__ __launch_bounds__(256) void hopf_epilogue_kernel(
    const float* __restrict__ pRe, const float* __restrict__ pIm,
    const float* __restrict__ z_re, const float* __restrict__ z_im,
    float* __restrict__ out)
{
    const int tid = blockIdx.x * blockDim.x + threadIdx.x;
    if (tid >= NN * UNITS) return;
    const int row = tid / UNITS;
    const int col = tid % UNITS;

    float re, im;
    if (col == 0) {
        re = z_re[row * UNITS];
        im = z_im[row * UNITS];
    } else {
        const int cc = col - 1;
        float sr = 0.f, si = 0.f;
        #pragma unroll
        for (int s = 0; s < SPLITS; ++s) {
            sr += pRe[(size_t)s * NN * 32 + row * 32 + cc];
            si += pIm[(size_t)s * NN * 32 + row * 32 + cc];
        }
        const float denom = coshf(2.f * sr) + cosf(2.f * si);
        re = sinhf(2.f * sr) / denom;
        im = sinf(2.f * si) / denom;
    }
    out[2 * tid]     = re;
    out[2 * tid + 1] = im;
}

extern "C" void kernel_launch(void* const* d_in, const int* in_sizes, int n_in,
                              void* d_out, int out_size, void* d_ws, size_t ws_size,
                              hipStream_t stream) {
    const float* x_re = (const float*)d_in[0];   // N x 32
    const float* x_im = (const float*)d_in[1];
    const float* z_re = (const float*)d_in[2];   // N x 33
    const float* z_im = (const float*)d_in[3];
    const float* A_re = (const float*)d_in[4];   // N x N
    const float* A_im = (const float*)d_in[5];
    const float* B_re = (const float*)d_in[6];
    const float* B_im = (const float*)d_in[7];

    float* pk  = (float*)d_ws;                               // 4 MB packed RHS
    float* pRe = pk + (size_t)(NN / 4) * 512;                // 8 MB partials
    float* pIm = pRe + (size_t)SPLITS * NN * 32;             // 8 MB partials

    hopf_pack_kernel<<<(2048 * 32) / 256, 256, 0, stream>>>(z_re, z_im, x_re, x_im, pk);

    hopf_gemm_kernel<<<512, 256, 0, stream>>>(A_re, A_im, B_re, B_im, pk, pRe, pIm);

    const int total = NN * UNITS;
    hopf_epilogue_kernel<<<(total + 255) / 256, 256, 0, stream>>>(pRe, pIm, z_re, z_im,
                                                                  (float*)d_out);
}